// CapacityBasedRouter_42700564857356
// MI455X (gfx1250) — compile-verified
//
#include <hip/hip_runtime.h>
#include <hip/hip_bf16.h>
#include <math.h>

typedef __attribute__((ext_vector_type(2))) float v2f;
typedef __attribute__((ext_vector_type(8))) float v8f;

#define N_TOKENS 8192
#define D_MODEL  4096
#define N_EXP    8
#define CAPACITY 1024
#define LB_W     0.01f
#define Z_W      0.001f

#define TPB1          256                      // 8 waves of 32
#define TOK_PER_WAVE  16
#define TOK_PER_BLOCK 128
#define NBLK1         (N_TOKENS / TOK_PER_BLOCK)   // 64
#define NSLOT         (2 * N_TOKENS)               // 16384

// ---------------- workspace layout (units of 4-byte elements) ----------------
#define WS_E     0          // int[16384]   flattened top-2 expert ids
#define WS_P     16384      // float[16384] flattened top-2 probs
#define WS_PART  32768      // float[64*9]  per-block partial sums
#define WS_CNT   33408      // int[8]       per-expert slot counts
#define WS_USAGE 33416      // int[8]       kept-top1 histogram
#define WS_ELIST 33536      // int[8*16384] per-expert slot-id lists

__global__ void k_init(int* __restrict__ cnt, int* __restrict__ usage) {
    int t = threadIdx.x;
    if (t < N_EXP) { cnt[t] = 0; usage[t] = 0; }
}

// GEMM (WMMA f32 16x16x4) + softmax + top2 + per-block reductions, fused.
// B columns 8..15 contain duplicated expert-0 data; the resulting garbage in
// D columns 8..15 is never read, so no masking (keeps the inner loop
// branch-free: 2x global_load_b64 + 1x v_wmma per k-step, EXEC all-ones).
__global__ void __launch_bounds__(TPB1)
k_gate(const float* __restrict__ x, const float* __restrict__ gw,
       int* __restrict__ ws_e, float* __restrict__ ws_p,
       float* __restrict__ part) {
    __shared__ float s_t[TOK_PER_BLOCK * 17];   // transposed logits (pad 17)
    __shared__ float s_red[TOK_PER_BLOCK * 9];  // reduction staging

    const int tid  = threadIdx.x;
    const int wave = tid >> 5;
    const int lane = tid & 31;
    const int half = lane >> 4;     // 0: K=0,1   1: K=2,3
    const int l16  = lane & 15;

    const int tok0 = blockIdx.x * TOK_PER_BLOCK + wave * TOK_PER_WAVE;

    // A fragment: row = tok0+l16, cols k+2*half, k+2*half+1
    const float* aptr = x + (size_t)(tok0 + l16) * D_MODEL + 2 * half;
    // B fragment: expert row = l16, clamped for lanes 8..15 (harmless dup)
    const int erow = (l16 < N_EXP) ? l16 : 0;
    const float* bptr = gw + (size_t)erow * D_MODEL + 2 * half;

    v8f acc = {0.f, 0.f, 0.f, 0.f, 0.f, 0.f, 0.f, 0.f};

#pragma unroll 8
    for (int k = 0; k < D_MODEL; k += 4) {
        v2f a, b;
        a.x = aptr[k];
        a.y = aptr[k + 1];
        b.x = bptr[k];
        b.y = bptr[k + 1];
        acc = __builtin_amdgcn_wmma_f32_16x16x4_f32(
                  false, a, false, b, (short)0, acc, false, false);
    }

    // C/D layout: VGPR r -> M = r + 8*half, N = l16. Transpose through LDS.
    {
        const int mbase = wave * TOK_PER_WAVE + 8 * half;
#pragma unroll
        for (int r = 0; r < 8; ++r)
            s_t[(mbase + r) * 17 + l16] = acc[r];
    }
    __syncthreads();

    // One thread per token: softmax(8), logsumexp, top-2 + renorm.
    float probs[N_EXP];
    if (tid < TOK_PER_BLOCK) {
        float lg[N_EXP];
#pragma unroll
        for (int e = 0; e < N_EXP; ++e) lg[e] = s_t[tid * 17 + e];

        float m = lg[0];
#pragma unroll
        for (int e = 1; e < N_EXP; ++e) m = fmaxf(m, lg[e]);
        float s = 0.f;
#pragma unroll
        for (int e = 0; e < N_EXP; ++e) { probs[e] = __expf(lg[e] - m); s += probs[e]; }
        const float inv = 1.f / s;
#pragma unroll
        for (int e = 0; e < N_EXP; ++e) probs[e] *= inv;
        const float lse = m + __logf(s);

        // top1: first strict max (matches top_k tie -> lowest index)
        int   i1 = 0;  float b1 = lg[0];
#pragma unroll
        for (int e = 1; e < N_EXP; ++e) if (lg[e] > b1) { b1 = lg[e]; i1 = e; }
        // top2: first strict max excluding i1
        int   i2 = -1; float b2 = -INFINITY;
#pragma unroll
        for (int e = 0; e < N_EXP; ++e)
            if (e != i1 && lg[e] > b2) { b2 = lg[e]; i2 = e; }

        const float mm = fmaxf(b1, b2);
        const float e1 = __expf(b1 - mm), e2 = __expf(b2 - mm);
        const float iss = 1.f / (e1 + e2);

        const int tok = blockIdx.x * TOK_PER_BLOCK + tid;
        ws_e[2 * tok]     = i1;
        ws_e[2 * tok + 1] = i2;
        ws_p[2 * tok]     = e1 * iss;
        ws_p[2 * tok + 1] = e2 * iss;

#pragma unroll
        for (int e = 0; e < N_EXP; ++e) s_red[tid * 9 + e] = probs[e];
        s_red[tid * 9 + 8] = lse * lse;
    }
    __syncthreads();

    // Fixed-order per-block reduction (deterministic): 9 columns.
    if (tid < 9) {
        float s = 0.f;
        for (int r = 0; r < TOK_PER_BLOCK; ++r) s += s_red[r * 9 + tid];
        part[blockIdx.x * 9 + tid] = s;
    }
}

// Bucket every (token,k) slot into its expert's list. List order is
// nondeterministic but downstream rank is an order-independent count.
__global__ void k_scatter(const int* __restrict__ ws_e,
                          int* __restrict__ cnt, int* __restrict__ elist) {
    const int i = blockIdx.x * 256 + threadIdx.x;
    if (i < NSLOT) {
        const int e   = ws_e[i];
        const int pos = atomicAdd(&cnt[e], 1);
        elist[e * NSLOT + pos] = i;
    }
}

// rank_i = #{j in same expert : p_j > p_i  or (p_j == p_i and j < i)}
// == stable lexsort((-p, e)) rank.  64 blocks per expert; wave scans its
// expert's list via broadcast loads, probs staged in LDS (64 KB).
__global__ void __launch_bounds__(256)
k_rank(const float* __restrict__ ws_p, const int* __restrict__ cnt,
       const int* __restrict__ elist, int* __restrict__ usage,
       float* __restrict__ out_idx, float* __restrict__ out_probs) {
    __shared__ float s_p[NSLOT];   // 64 KB

    const int e   = blockIdx.x >> 6;                       // expert
    const int pos = (blockIdx.x & 63) * 256 + threadIdx.x; // list position
    const int n   = cnt[e];
    if ((blockIdx.x & 63) * 256 >= n) return;              // uniform early-out

    for (int j = threadIdx.x; j < NSLOT; j += 256) s_p[j] = ws_p[j];
    __syncthreads();

    if (pos >= n) return;
    const int*  lst = elist + e * NSLOT;
    const int   i   = lst[pos];
    const float p   = s_p[i];

    int rank = 0;
    for (int q = 0; q < n; ++q) {
        const int   j  = lst[q];
        const float pj = s_p[j];
        rank += (pj > p) || (pj == p && j < i);
    }
    const bool keep = rank < CAPACITY;
    out_idx[i]   = keep ? (float)e : -1.0f;
    out_probs[i] = keep ? p : 0.0f;
    if (((i & 1) == 0) && keep) atomicAdd(&usage[e], 1);   // kept top-1
}

// Final fixed-order reduce: lb, zl, usage -> d_out tail.
__global__ void k_final(const float* __restrict__ part,
                        const int* __restrict__ usage,
                        float* __restrict__ out) {
    __shared__ float s_ps[9];
    const int t = threadIdx.x;
    if (t < 9) {
        float s = 0.f;
        for (int b = 0; b < NBLK1; ++b) s += part[b * 9 + t];
        s_ps[t] = s;
    }
    __syncthreads();
    if (t == 0) {
        float lb = 0.f;
        for (int e = 0; e < N_EXP; ++e) lb += s_ps[e] * (float)usage[e];
        lb *= LB_W / ((float)N_TOKENS * (float)N_EXP);
        out[2 * NSLOT]     = lb;
        out[2 * NSLOT + 1] = s_ps[8] / (float)N_TOKENS * Z_W;
    }
    if (t < N_EXP) out[2 * NSLOT + 2 + t] = (float)usage[t];
}

extern "C" void kernel_launch(void* const* d_in, const int* in_sizes, int n_in,
                              void* d_out, int out_size, void* d_ws, size_t ws_size,
                              hipStream_t stream) {
    const float* x  = (const float*)d_in[0];   // [8192, 4096]
    const float* gw = (const float*)d_in[1];   // [8, 4096]
    float* out = (float*)d_out;                // 16384 idx | 16384 probs | lb | zl | usage[8]

    int*   wsi   = (int*)d_ws;
    float* wsf   = (float*)d_ws;
    int*   ws_e  = wsi + WS_E;
    float* ws_p  = wsf + WS_P;
    float* part  = wsf + WS_PART;
    int*   cnt   = wsi + WS_CNT;
    int*   usage = wsi + WS_USAGE;
    int*   elist = wsi + WS_ELIST;

    k_init   <<<1, 32, 0, stream>>>(cnt, usage);
    k_gate   <<<NBLK1, TPB1, 0, stream>>>(x, gw, ws_e, ws_p, part);
    k_scatter<<<(NSLOT + 255) / 256, 256, 0, stream>>>(ws_e, cnt, elist);
    k_rank   <<<N_EXP * 64, 256, 0, stream>>>(ws_p, cnt, elist, usage,
                                              out, out + NSLOT);
    k_final  <<<1, 64, 0, stream>>>(part, usage, out);
}